// DualHeadGAT_84258668413092
// MI455X (gfx1250) — compile-verified
//
#include <hip/hip_runtime.h>
#include <hip/hip_bf16.h>
#include <hip/hip_fp16.h>
#include <math.h>

typedef float v2f __attribute__((ext_vector_type(2)));
typedef float v8f __attribute__((ext_vector_type(8)));

#define NEG_SLOPE 0.2f

// ---- monotonic float <-> uint mapping so atomicMax(u32) orders floats ----
__device__ __forceinline__ unsigned f2key(float f) {
  unsigned b = __float_as_uint(f);
  return (b & 0x80000000u) ? ~b : (b | 0x80000000u);
}
__device__ __forceinline__ float key2f(unsigned k) {
  unsigned b = (k & 0x80000000u) ? (k ^ 0x80000000u) : ~k;
  return __uint_as_float(b);
}
__device__ __forceinline__ float elu_f(float v) { return v > 0.f ? v : expf(v) - 1.f; }

// ---- CDNA5 async global->LDS copy (ASYNCcnt path, §10 / CDNA5 ISA) ----
// VDST VGPR holds the LDS byte address, VADDR the 64-bit global address (GV mode).
__device__ __forceinline__ void async_g2l_b128(unsigned lds_off, const float* gsrc) {
  asm volatile("global_load_async_to_lds_b128 %0, %1, off"
               :
               : "v"(lds_off), "v"((unsigned long long)(const void*)gsrc)
               : "memory");
}
__device__ __forceinline__ void wait_asynccnt0() {
#if __has_builtin(__builtin_amdgcn_s_wait_asynccnt)
  __builtin_amdgcn_s_wait_asynccnt(0);
#else
  asm volatile("s_wait_asynccnt 0x0" ::: "memory");
#endif
}

// ---------------------------------------------------------------------------
// C[M,NCOL] = A[M,K] * B[K,NCOL], fp32, via V_WMMA_F32_16X16X4_F32.
// One block: 16 rows x NCOL cols. K tiled by 32, A/B staged in LDS through
// async b128 copies. Each wave owns TPW 16x16 column tiles.
// Launch: blockDim = 32 * (NCOL/16/TPW), grid = M/16 (M % 16 == 0).
// ---------------------------------------------------------------------------
template <int NCOL, int TPW>
__global__ void k_gemm_lds(const float* __restrict__ A, const float* __restrict__ B,
                           float* __restrict__ C, int K)
{
  constexpr int AST = 36;                 // sA row stride (16B-aligned, depads banks)
  __shared__ float sA[16 * AST];
  __shared__ float sB[32 * NCOL];

  const int tid  = threadIdx.x;
  const int lane = tid & 31;
  const int w    = tid >> 5;
  const int rowBase = blockIdx.x * 16;

  // fp32 WMMA lane layout: lanes 0-15 -> K={0,1}, lanes 16-31 -> K={2,3}
  const int r    = lane & 15;
  const int kk   = (lane >> 4) << 1;
  const int mrow = (lane >> 4) << 3;      // C: lanes 16-31 hold M+8

  const unsigned sA_base = (unsigned)(unsigned long long)(const void*)&sA[0];
  const unsigned sB_base = (unsigned)(unsigned long long)(const void*)&sB[0];

  v8f acc[TPW] = {};

  for (int k0 = 0; k0 < K; k0 += 32) {
    // stage A tile: 16 rows x 32 floats = 128 b128 chunks
    for (int c = tid; c < 16 * 8; c += blockDim.x) {
      const int rr = c >> 3, kq = (c & 7) << 2;
      async_g2l_b128(sA_base + (unsigned)(rr * AST + kq) * 4u,
                     A + (size_t)(rowBase + rr) * K + k0 + kq);
    }
    // stage B tile: 32 rows x NCOL floats = 8*NCOL b128 chunks
    for (int c = tid; c < 8 * NCOL; c += blockDim.x) {
      const int rr = c / (NCOL / 4), cq = (c % (NCOL / 4)) << 2;
      async_g2l_b128(sB_base + (unsigned)(rr * NCOL + cq) * 4u,
                     B + (size_t)(k0 + rr) * NCOL + cq);
    }
    wait_asynccnt0();
    __syncthreads();

#pragma unroll
    for (int kc = 0; kc < 32; kc += 4) {
      v2f a;
      a.x = sA[r * AST + kc + kk];
      a.y = sA[r * AST + kc + kk + 1];
#pragma unroll
      for (int i = 0; i < TPW; ++i) {
        const int col = (w * TPW + i) * 16 + r;
        v2f b;
        b.x = sB[(kc + kk) * NCOL + col];
        b.y = sB[(kc + kk + 1) * NCOL + col];
        acc[i] = __builtin_amdgcn_wmma_f32_16x16x4_f32(false, a, false, b, (short)0,
                                                       acc[i], false, false);
      }
    }
    __syncthreads();
  }

#pragma unroll
  for (int i = 0; i < TPW; ++i) {
    const int col = (w * TPW + i) * 16 + r;
#pragma unroll
    for (int j = 0; j < 8; ++j)
      C[(size_t)(rowBase + mrow + j) * NCOL + col] = acc[i][j];
  }
}

__global__ void k_zero(float* __restrict__ p, size_t n)
{
  size_t i = (size_t)blockIdx.x * blockDim.x + threadIdx.x;
  const size_t stride = (size_t)gridDim.x * blockDim.x;
  for (; i < n; i += stride) p[i] = 0.f;
}

// a_src[n,h] = sum_c h[n,h,c]*att_src[h,c]; likewise a_dst. One thread per (n,h).
__global__ void k_attn(const float* __restrict__ hmat, const float* __restrict__ atts,
                       const float* __restrict__ attd, float* __restrict__ as,
                       float* __restrict__ ad, int N, int H, int C)
{
  const int i = blockIdx.x * blockDim.x + threadIdx.x;
  if (i >= N * H) return;
  const int h = i % H;
  const float* hp = hmat + (size_t)i * C;   // n*H*C + h*C == i*C
  float sa = 0.f, sd = 0.f;
  for (int c = 0; c < C; ++c) {
    const float v = hp[c];
    sa += v * atts[h * C + c];
    sd += v * attd[h * C + c];
  }
  as[i] = sa;
  ad[i] = sd;
}

template <int H>
__global__ void k_edge_max(const int* __restrict__ ei, int E, int Et,
                           const float* __restrict__ asrc, const float* __restrict__ adst,
                           float* __restrict__ raw, unsigned* __restrict__ emaxk)
{
  const int e = blockIdx.x * blockDim.x + threadIdx.x;
  if (e >= Et) return;
  const int s = (e < E) ? ei[e]     : (e - E);
  const int d = (e < E) ? ei[E + e] : (e - E);
#pragma unroll
  for (int h = 0; h < H; ++h) {
    float x = asrc[s * H + h] + adst[d * H + h];
    x = (x > 0.f) ? x : NEG_SLOPE * x;
    raw[(size_t)e * H + h] = x;
    atomicMax(&emaxk[d * H + h], f2key(x));
  }
}

template <int H>
__global__ void k_edge_exp(const int* __restrict__ ei, int E, int Et,
                           const unsigned* __restrict__ emaxk,
                           float* __restrict__ eexp, float* __restrict__ denom)
{
  const int e = blockIdx.x * blockDim.x + threadIdx.x;
  if (e >= Et) return;
  const int d = (e < E) ? ei[E + e] : (e - E);
#pragma unroll
  for (int h = 0; h < H; ++h) {
    const float w = expf(eexp[(size_t)e * H + h] - key2f(emaxk[d * H + h]));
    eexp[(size_t)e * H + h] = w;
    atomicAdd(&denom[d * H + h], w);
  }
}

template <int H>
__global__ void k_edge_alpha(const int* __restrict__ ei, int E, int Et,
                             const float* __restrict__ denom, float* __restrict__ eexp)
{
  const int e = blockIdx.x * blockDim.x + threadIdx.x;
  if (e >= Et) return;
  const int d = (e < E) ? ei[E + e] : (e - E);
#pragma unroll
  for (int h = 0; h < H; ++h)
    eexp[(size_t)e * H + h] /= (denom[d * H + h] + 1e-16f);
}

// layer-1 message scatter: thread = (edge, channel); 4 heads per thread
__global__ void k_scatter1(const int* __restrict__ ei, int E, int Et,
                           const float* __restrict__ h1, const float* __restrict__ alpha,
                           float* __restrict__ acc)
{
  const long long idx = (long long)blockIdx.x * blockDim.x + threadIdx.x;
  if (idx >= (long long)Et * 64) return;
  const int e = (int)(idx >> 6);
  const int c = (int)(idx & 63);
  const int s = (e < E) ? ei[e]     : (e - E);
  const int d = (e < E) ? ei[E + e] : (e - E);
#pragma unroll
  for (int h = 0; h < 4; ++h)
    atomicAdd(&acc[(size_t)d * 256 + h * 64 + c],
              h1[(size_t)s * 256 + h * 64 + c] * alpha[(size_t)e * 4 + h]);
}

// layer-2 message scatter: single head, 64 channels
__global__ void k_scatter2(const int* __restrict__ ei, int E, int Et,
                           const float* __restrict__ h2, const float* __restrict__ alpha,
                           float* __restrict__ acc)
{
  const long long idx = (long long)blockIdx.x * blockDim.x + threadIdx.x;
  if (idx >= (long long)Et * 64) return;
  const int e = (int)(idx >> 6);
  const int c = (int)(idx & 63);
  const int s = (e < E) ? ei[e]     : (e - E);
  const int d = (e < E) ? ei[E + e] : (e - E);
  atomicAdd(&acc[(size_t)d * 64 + c], h2[(size_t)s * 64 + c] * alpha[e]);
}

__global__ void k_bias_elu(float* __restrict__ a, const float* __restrict__ bias,
                           size_t n, int F)
{
  const size_t i = (size_t)blockIdx.x * blockDim.x + threadIdx.x;
  if (i >= n) return;
  a[i] = elu_f(a[i] + bias[i % F]);
}

// out node features (ELU) + per-node score; one 64-thread block per node
__global__ void k_final(const float* __restrict__ acc2, const float* __restrict__ b2,
                        const float* __restrict__ Ws, const float* __restrict__ bs,
                        float* __restrict__ out, int N)
{
  __shared__ float red[64];
  const int n = blockIdx.x;
  const int c = threadIdx.x;
  float v = acc2[(size_t)n * 64 + c] + b2[c];
  v = elu_f(v);
  out[(size_t)n * 64 + c] = v;
  red[c] = v * Ws[c];
  __syncthreads();
  for (int s = 32; s > 0; s >>= 1) {
    if (c < s) red[c] += red[c + s];
    __syncthreads();
  }
  if (c == 0) out[(size_t)N * 64 + n] = red[0] + bs[0];
}

extern "C" void kernel_launch(void* const* d_in, const int* in_sizes, int n_in,
                              void* d_out, int out_size, void* d_ws, size_t ws_size,
                              hipStream_t stream)
{
  const float* x   = (const float*)d_in[0];
  const int*   ei  = (const int*)d_in[1];
  const float* W1  = (const float*)d_in[2];
  const float* as1 = (const float*)d_in[3];
  const float* ad1 = (const float*)d_in[4];
  const float* b1  = (const float*)d_in[5];
  const float* W2  = (const float*)d_in[6];
  const float* as2 = (const float*)d_in[7];
  const float* ad2 = (const float*)d_in[8];
  const float* b2  = (const float*)d_in[9];
  const float* Ws  = (const float*)d_in[10];
  const float* bs  = (const float*)d_in[11];
  float* out = (float*)d_out;

  const int IN = 128, F1 = 256, H1 = 4, C = 64;
  const int N  = in_sizes[0] / IN;
  const int E  = in_sizes[1] / 2;
  const int Et = E + N;

  // workspace layout (zeroed region is contiguous, starting at acc1)
  float* f = (float*)d_ws;
  float*    h1   = f;            f += (size_t)N * F1;
  float*    acc1 = f;            f += (size_t)N * F1;  // zero
  float*    den1 = f;            f += (size_t)N * H1;  // zero
  unsigned* em1  = (unsigned*)f; f += (size_t)N * H1;  // zero
  float*    acc2 = f;            f += (size_t)N * C;   // zero
  float*    den2 = f;            f += (size_t)N;       // zero
  unsigned* em2  = (unsigned*)f; f += (size_t)N;       // zero
  float*    asr1 = f;            f += (size_t)N * H1;
  float*    adt1 = f;            f += (size_t)N * H1;
  float*    ee1  = f;            f += (size_t)Et * H1;
  float*    h2   = f;            f += (size_t)N * C;
  float*    asr2 = f;            f += (size_t)N;
  float*    adt2 = f;            f += (size_t)N;
  float*    ee2  = f;            f += (size_t)Et;
  (void)ws_size; (void)n_in; (void)out_size;

  const size_t zeroN = (size_t)N * (F1 + H1 + H1 + C + 1 + 1);
  k_zero<<<2048, 256, 0, stream>>>(acc1, zeroN);

  const int eb = (Et + 255) / 256;
  const long long scN = (long long)Et * 64;
  const int sb = (int)((scN + 255) / 256);

  // ---- layer 1 ----
  k_gemm_lds<256, 2><<<N / 16, 256, 0, stream>>>(x, W1, h1, IN);
  k_attn<<<(N * H1 + 255) / 256, 256, 0, stream>>>(h1, as1, ad1, asr1, adt1, N, H1, C);
  k_edge_max<4><<<eb, 256, 0, stream>>>(ei, E, Et, asr1, adt1, ee1, em1);
  k_edge_exp<4><<<eb, 256, 0, stream>>>(ei, E, Et, em1, ee1, den1);
  k_edge_alpha<4><<<eb, 256, 0, stream>>>(ei, E, Et, den1, ee1);
  k_scatter1<<<sb, 256, 0, stream>>>(ei, E, Et, h1, ee1, acc1);
  k_bias_elu<<<(int)(((size_t)N * F1 + 255) / 256), 256, 0, stream>>>(acc1, b1,
                                                                      (size_t)N * F1, F1);

  // ---- layer 2 ----
  k_gemm_lds<64, 1><<<N / 16, 128, 0, stream>>>(acc1, W2, h2, F1);
  k_attn<<<(N + 255) / 256, 256, 0, stream>>>(h2, as2, ad2, asr2, adt2, N, 1, C);
  k_edge_max<1><<<eb, 256, 0, stream>>>(ei, E, Et, asr2, adt2, ee2, em2);
  k_edge_exp<1><<<eb, 256, 0, stream>>>(ei, E, Et, em2, ee2, den2);
  k_edge_alpha<1><<<eb, 256, 0, stream>>>(ei, E, Et, den2, ee2);
  k_scatter2<<<sb, 256, 0, stream>>>(ei, E, Et, h2, ee2, acc2);
  k_final<<<N, 64, 0, stream>>>(acc2, b2, Ws, bs, out, N);
}